// MihGNNEmbedding4_79216376807934
// MI455X (gfx1250) — compile-verified
//
#include <hip/hip_runtime.h>

#define N_NODES 20000
#define DIM     128
#define KNBR    16
#define STEPS   2
#define LAYERS  2
#define NEDGE   8192
#define XPITCH  132   // 128 + 4 dword pad -> conflict-free LDS column reads

typedef __attribute__((ext_vector_type(2))) float v2f;
typedef __attribute__((ext_vector_type(8))) float v8f;

__device__ __forceinline__ float wave_reduce_add(float v) {
#pragma unroll
  for (int off = 16; off > 0; off >>= 1) v += __shfl_xor(v, off, 32);
  return v;
}

__global__ void zero_out_kernel(float* out) {
  if (threadIdx.x == 0 && blockIdx.x == 0) out[0] = 0.0f;
}

// One wave32 per node. Lane l owns float4 slice d in [4l, 4l+4).
__global__ __launch_bounds__(256) void attention_kernel(
    const float* __restrict__ emb, const int* __restrict__ neighbors,
    const float* __restrict__ fc_w, const float* __restrict__ fc_b_p,
    float* __restrict__ att) {
  const int lane = threadIdx.x & 31;
  const int wv   = threadIdx.x >> 5;
  const int n    = blockIdx.x * 8 + wv;
  if (n >= N_NODES) return;

  const float  fcb = fc_b_p[0];
  const float4 wn  = ((const float4*)fc_w)[lane];            // fc_w[:D]
  const float4 wsv = ((const float4*)(fc_w + DIM))[lane];    // fc_w[D:]
  const float4 es  = ((const float4*)(emb + (size_t)n * DIM))[lane];

  const float self_score =
      wave_reduce_add(es.x * wsv.x + es.y * wsv.y + es.z * wsv.z + es.w * wsv.w);

  float4 acc = {0.f, 0.f, 0.f, 0.f};

  for (int s = 0; s < STEPS; ++s) {
    const int* nbr = neighbors + ((size_t)s * N_NODES + n) * KNBR;
    float4 ne[KNBR];
    float  sc[KNBR];
#pragma unroll
    for (int k = 0; k < KNBR; ++k) {
      const int idx = nbr[k];  // wave-uniform -> scalar load
      ne[k] = ((const float4*)(emb + (size_t)idx * DIM))[lane];
      float d = ne[k].x * wn.x + ne[k].y * wn.y + ne[k].z * wn.z + ne[k].w * wn.w;
      float t = wave_reduce_add(d) + self_score + fcb;
      sc[k] = (t >= 0.f) ? t : 0.2f * t;  // leaky relu
    }
    float m = sc[0];
#pragma unroll
    for (int k = 1; k < KNBR; ++k) m = fmaxf(m, sc[k]);
    float ssum = 0.f;
#pragma unroll
    for (int k = 0; k < KNBR; ++k) { sc[k] = __expf(sc[k] - m); ssum += sc[k]; }
    const float inv = 1.f / ssum;
#pragma unroll
    for (int k = 0; k < KNBR; ++k) {
      const float a = sc[k] * inv;
      acc.x += a * ne[k].x; acc.y += a * ne[k].y;
      acc.z += a * ne[k].z; acc.w += a * ne[k].w;
    }
  }
  const float kf = (float)(STEPS * KNBR);  // each step adds K*emb
  acc.x += kf * es.x; acc.y += kf * es.y; acc.z += kf * es.z; acc.w += kf * es.w;
  ((float4*)(att + (size_t)n * DIM))[lane] = acc;
}

// One block per 16 edges; wave w handles feature tile f0 = 16*w.
// out[e,f] = sum_d x[e,d] * W[l][f][d]  via V_WMMA_F32_16X16X4_F32 chains.
__global__ __launch_bounds__(256) void edge_loss_kernel(
    const int* __restrict__ edges, const int* __restrict__ labels,
    const float* __restrict__ emb, const float* __restrict__ att,
    const float* __restrict__ layer_w, const float* __restrict__ layer_b,
    float* __restrict__ out) {
  __shared__ float Xs[16 * XPITCH];
  __shared__ float Xd[16 * XPITCH];
  __shared__ float diff_lds[16];

  const int tid = threadIdx.x;
  const int eb  = blockIdx.x * 16;
  if (tid < 16) diff_lds[tid] = 0.f;

  // Stage 16 src rows + 16 dst rows (att or emb, per label) into LDS.
#pragma unroll
  for (int it = 0; it < 2; ++it) {
    const int i  = tid + it * 256;   // 512 float4 slots per side
    const int e  = i >> 5;
    const int c  = i & 31;
    const int ge = eb + e;
    const float* base = (labels[ge] == 1) ? att : emb;
    const int si = edges[2 * ge];
    const int di = edges[2 * ge + 1];
    const float4 vs = ((const float4*)(base + (size_t)si * DIM))[c];
    const float4 vd = ((const float4*)(base + (size_t)di * DIM))[c];
    *(float4*)&Xs[e * XPITCH + c * 4] = vs;
    *(float4*)&Xd[e * XPITCH + c * 4] = vd;
  }
  __syncthreads();

  const int lane = tid & 31;
  const int wv   = tid >> 5;
  const int half = lane >> 4;   // A/B K-halves: lanes 0-15 -> K=0,1; 16-31 -> K=2,3
  const int lr   = lane & 15;   // M (edge) / N (feature) index within tile
  const int f0   = wv * 16;

  v8f sumS = {0.f,0.f,0.f,0.f,0.f,0.f,0.f,0.f};
  v8f sumD = {0.f,0.f,0.f,0.f,0.f,0.f,0.f,0.f};

  for (int l = 0; l < LAYERS; ++l) {
    const float* Wrow = layer_w + ((size_t)l * DIM + f0 + lr) * DIM;
    const float  bias = layer_b[l * DIM + f0 + lr];

    v2f B[32];
#pragma unroll
    for (int kk = 0; kk < 32; ++kk)
      B[kk] = *(const v2f*)(Wrow + 4 * kk + 2 * half);   // B[k][n] = W[f0+n][kbase+k]

#pragma unroll
    for (int side = 0; side < 2; ++side) {
      const float* X = side ? Xd : Xs;
      v8f c8 = {0.f,0.f,0.f,0.f,0.f,0.f,0.f,0.f};
#pragma unroll
      for (int kk = 0; kk < 32; ++kk) {
        const v2f a2 = *(const v2f*)(X + lr * XPITCH + 4 * kk + 2 * half);
        c8 = __builtin_amdgcn_wmma_f32_16x16x4_f32(
            false, a2, false, B[kk], (short)0, c8, false, false);
      }
      v8f* sum = side ? &sumD : &sumS;
#pragma unroll
      for (int j = 0; j < 8; ++j) {
        const float t = c8[j] + bias;
        (*sum)[j] += (t > 0.f) ? t : 0.f;   // relu, then sum over layers
      }
    }
  }

  // Per-lane squared diff; C element (vgpr j, lane) is edge m = j + 8*half, f = f0 + lr.
  float d2[8];
#pragma unroll
  for (int j = 0; j < 8; ++j) { const float t = sumS[j] - sumD[j]; d2[j] = t * t; }
#pragma unroll
  for (int j = 0; j < 8; ++j) {
#pragma unroll
    for (int off = 1; off < 16; off <<= 1) d2[j] += __shfl_xor(d2[j], off, 32);
  }
  if (lr == 0) {
#pragma unroll
    for (int j = 0; j < 8; ++j) atomicAdd(&diff_lds[j + 8 * half], d2[j]);
  }
  __syncthreads();

  float lv = 0.f;
  if (tid < 16) {
    const float df  = diff_lds[tid] * (1.0f / (float)DIM);
    const float p   = __expf(-df);
    const float lab = (float)labels[eb + tid];
    const float r   = lab - p;
    lv = 0.5f * r * r;
  }
  if (wv == 0) {
    lv = wave_reduce_add(lv);
    if (lane == 0) atomicAdd(out, lv);
  }
}

extern "C" void kernel_launch(void* const* d_in, const int* in_sizes, int n_in,
                              void* d_out, int out_size, void* d_ws, size_t ws_size,
                              hipStream_t stream) {
  const int*   edges      = (const int*)d_in[0];
  const int*   labels     = (const int*)d_in[1];
  const int*   neighbors  = (const int*)d_in[2];
  const float* embeddings = (const float*)d_in[3];
  const float* fc_w       = (const float*)d_in[4];
  const float* fc_b       = (const float*)d_in[5];
  const float* layer_w    = (const float*)d_in[6];
  const float* layer_b    = (const float*)d_in[7];
  float* out = (float*)d_out;
  float* att = (float*)d_ws;  // N*DIM floats = 10.24 MB scratch

  zero_out_kernel<<<1, 64, 0, stream>>>(out);
  attention_kernel<<<(N_NODES + 7) / 8, 256, 0, stream>>>(embeddings, neighbors,
                                                          fc_w, fc_b, att);
  edge_loss_kernel<<<NEDGE / 16, 256, 0, stream>>>(edges, labels, embeddings, att,
                                                   layer_w, layer_b, out);
}